// FragmentPositionDistribution3_64802466562899
// MI455X (gfx1250) — compile-verified
//
#include <hip/hip_runtime.h>

// ---------------------------------------------------------------------------
// FragmentPositionDistribution3 for MI455X (gfx1250, wave32).
//
// Pipeline (all on `stream`, stream-ordered):
//   K0: 1 thread      -> precompute 6 spline/outside constants into ws
//   K1: 256 blocks    -> LSE[roi][cluster] = logsumexp_b(baseline+delta),
//                        cross-lane sum reduction done on the matrix engine
//                        via V_WMMA_F32_16X16X4_F32 with an all-ones B.
//   K2: 8M threads    -> lp_pos = baseline[g]+delta[g]-LSE, lp_size = spline.
//                        Pure streaming + L2-resident gathers (memory bound:
//                        ~24 B/fragment streamed => ~8 us at 23.3 TB/s).
// ---------------------------------------------------------------------------

typedef __attribute__((ext_vector_type(2))) float v2f;
typedef __attribute__((ext_vector_type(8))) float v8f;

#define F_BINW    1000
#define F_NCLUST  32
#define F_NROI    256
#define F_WIN_LO  (-100000)
#define F_BINSIZE 200
#define F_SPLINEW 1024
#define F_L0      0.33333334f   // cumsum(1/3) in f32
#define F_L1      0.66666669f

// --- K0: spline constants --------------------------------------------------
// sp[0..3] = normalized knot heights, sp[4] = logprob_inside,
// sp[5] = log1p(-exp(lpi)) + lpi (full "outside" value)
__global__ void spline_precompute_kernel(const float* __restrict__ uh,
                                         const float* __restrict__ lpi_p,
                                         float* __restrict__ sp) {
  if (blockIdx.x == 0 && threadIdx.x == 0) {
    float h0 = expf(uh[0]), h1 = expf(uh[1]), h2 = expf(uh[2]), h3 = expf(uh[3]);
    float area = (0.5f * (h0 + h1) + 0.5f * (h1 + h2) + 0.5f * (h2 + h3)) * 0.33333334f;
    float inv = 1.0f / area;
    sp[0] = h0 * inv; sp[1] = h1 * inv; sp[2] = h2 * inv; sp[3] = h3 * inv;
    float lpi = lpi_p[0];
    sp[4] = lpi;
    sp[5] = log1pf(-expf(lpi)) + lpi;
  }
}

// --- K1: per-(region_oi, cluster) logsumexp over 1000 bins -----------------
// Block = 1024 threads = 32 waves; wave w == cluster w of region blockIdx.x.
__global__ __launch_bounds__(1024)
void lse_kernel(const float* __restrict__ baseline,   // [2000,1000]
                const float* __restrict__ delta,      // [2000,32,1000]
                const int*   __restrict__ regions_oi, // [256]
                float*       __restrict__ lse_out)    // [256,32]
{
  __shared__ float sBase[F_BINW];
  __shared__ float sPart[F_NCLUST][33];   // [cluster][lane], padded
  __shared__ float sMax[F_NCLUST];

  const int roi = blockIdx.x;
  const int reg = regions_oi[roi];
  const int tid = threadIdx.x;

  for (int b = tid; b < F_BINW; b += 1024)
    sBase[b] = baseline[(size_t)reg * F_BINW + b];
  __syncthreads();

  const int wave = tid >> 5;   // cluster 0..31 (wave-uniform)
  const int lane = tid & 31;
  const float* __restrict__ drow =
      delta + ((size_t)reg * F_NCLUST + wave) * F_BINW;

  // pass 1: row max (wave32 butterfly reduce)
  float m = -1e30f;
  for (int b = lane; b < F_BINW; b += 32)
    m = fmaxf(m, sBase[b] + drow[b]);
#pragma unroll
  for (int off = 16; off > 0; off >>= 1)
    m = fmaxf(m, __shfl_xor(m, off, 32));
  if (lane == 0) sMax[wave] = m;

  // pass 2: per-lane partial sums of exp(x - max)  (delta re-read hits L2)
  float s = 0.0f;
  for (int b = lane; b < F_BINW; b += 32)
    s += expf((sBase[b] + drow[b]) - m);
  sPart[wave][lane] = s;
  __syncthreads();

  // Matrix-engine cross-lane reduction: waves 0/1 sum rows 0-15 / 16-31 of
  // the 32x32 partial matrix with V_WMMA_F32_16X16X4_F32, B = ones(4x16).
  if (wave < 2) {                       // wave-uniform branch: EXEC stays full
    const int rowBase = wave << 4;
    const int m16   = lane & 15;
    const int khalf = (lane >> 4) << 1; // lanes 0-15 -> K0,K1; 16-31 -> K2,K3
    v2f bones; bones[0] = 1.0f; bones[1] = 1.0f;
    v8f acc = {};
#pragma unroll
    for (int k0 = 0; k0 < 32; k0 += 4) {
      v2f a;
      a[0] = sPart[rowBase + m16][k0 + khalf];
      a[1] = sPart[rowBase + m16][k0 + khalf + 1];
      acc = __builtin_amdgcn_wmma_f32_16x16x4_f32(
          /*neg_a=*/false, a, /*neg_b=*/false, bones,
          /*c_mod=*/(short)0, acc, /*reuse_a=*/false, /*reuse_b=*/false);
    }
    // D layout: lane L, vgpr j -> row (j + 8*(L>=16)), col L&15 (all cols eq).
    if ((lane & 15) == 0) {
      const int rbase = rowBase + ((lane >> 4) << 3);
#pragma unroll
      for (int j = 0; j < 8; ++j)
        lse_out[roi * F_NCLUST + rbase + j] = sMax[rbase + j] + logf(acc[j]);
    }
  }
}

// --- K2: the hot memory-bound per-fragment kernel --------------------------
__global__ __launch_bounds__(256)
void frag_kernel(const float* __restrict__ baseline,
                 const float* __restrict__ delta,
                 const int*   __restrict__ coords,      // [F,2]
                 const int*   __restrict__ regions_oi,  // [256]
                 const int*   __restrict__ lrix,        // [F]
                 const int*   __restrict__ lcix,        // [F]
                 const int*   __restrict__ labels,      // [100k]
                 const float* __restrict__ lse,         // [256,32]
                 const float* __restrict__ sp,          // [6]
                 float*       __restrict__ out,         // [F,2]
                 int F)
{
  const int f = blockIdx.x * blockDim.x + threadIdx.x;
  if (f >= F) return;

  const int2 c = reinterpret_cast<const int2*>(coords)[f];  // coalesced b64
  const int  r   = lrix[f];
  const int  lab = labels[lcix[f]];
  const int  reg = regions_oi[r];
  const unsigned bin = (unsigned)(c.x - F_WIN_LO) / (unsigned)F_BINSIZE;

  // two L2-resident 4B gathers + one 32KB-table lookup
  const float lp_pos = baseline[reg * F_BINW + bin]
                     + delta[((size_t)reg * F_NCLUST + lab) * F_BINW + bin]
                     - lse[r * F_NCLUST + lab];

  // spline constants (uniform addresses -> SMEM loads)
  const float s0 = sp[0], s1 = sp[1], s2 = sp[2], s3 = sp[3];
  const float lpi = sp[4], outsideVal = sp[5];

  const int   fs = c.y - c.x;                     // fragment size, >= 1
  const float x  = fminf((float)fs * (1.0f / (float)F_SPLINEW), 1.0f);
  int b = (x >= F_L0) + (x >= F_L1) + (x >= 1.0f);
  b = (b > 2) ? 2 : b;                            // searchsorted(right), clipped
  const float kl  = (b == 0) ? 0.0f : ((b == 1) ? F_L0 : F_L1);
  const float hb  = (b == 0) ? s0   : ((b == 1) ? s1   : s2);
  const float hb1 = (b == 0) ? s1   : ((b == 1) ? s2   : s3);
  const float alpha = (x - kl) * 3.0f;            // / widths[b] (=1/3)
  const float dens  = hb * (1.0f - alpha) + hb1 * alpha;
  const float lp_size = (fs > F_SPLINEW) ? outsideVal : (logf(dens) + lpi);

  float2 o; o.x = lp_pos; o.y = lp_size;
  reinterpret_cast<float2*>(out)[f] = o;          // coalesced b64 store
}

// ---------------------------------------------------------------------------
extern "C" void kernel_launch(void* const* d_in, const int* in_sizes, int n_in,
                              void* d_out, int out_size, void* d_ws, size_t ws_size,
                              hipStream_t stream) {
  const float* baseline   = (const float*)d_in[0];  // [2000,1000]
  const float* delta      = (const float*)d_in[1];  // [2000,32,1000]
  const float* spline_uh  = (const float*)d_in[2];  // [4]
  const float* lpi        = (const float*)d_in[3];  // [1]
  const int*   coords     = (const int*)d_in[4];    // [F,2]
  const int*   regions_oi = (const int*)d_in[5];    // [256]
  const int*   lrix       = (const int*)d_in[6];    // [F]
  const int*   lcix       = (const int*)d_in[7];    // [F]
  const int*   labels     = (const int*)d_in[8];    // [100k]

  float* wsF = (float*)d_ws;
  float* lse = wsF;                    // 256*32 = 8192 floats
  float* sp  = wsF + F_NROI * F_NCLUST; // 6 floats

  const int F = in_sizes[6];

  spline_precompute_kernel<<<1, 32, 0, stream>>>(spline_uh, lpi, sp);
  lse_kernel<<<F_NROI, 1024, 0, stream>>>(baseline, delta, regions_oi, lse);
  const int nb = (F + 255) / 256;
  frag_kernel<<<nb, 256, 0, stream>>>(baseline, delta, coords, regions_oi,
                                      lrix, lcix, labels, lse, sp,
                                      (float*)d_out, F);
}